// MultiHeadAttention_23854248362769
// MI455X (gfx1250) — compile-verified
//
#include <hip/hip_runtime.h>
#include <hip/hip_bf16.h>

// ---------------------------------------------------------------------------
// MHA forward for MI455X (gfx1250), fp32 end-to-end using V_WMMA_F32_16X16X4_F32.
// B=4, S=2048, D=1024, H=16, HD=64.  SCALE = sqrt(64) = 8.
// ---------------------------------------------------------------------------

typedef float v2f __attribute__((ext_vector_type(2)));
typedef float v8f __attribute__((ext_vector_type(8)));

__device__ __forceinline__ v8f wmma_f32(v2f a, v2f b, v8f c) {
    // (neg_a, A, neg_b, B, c_mod, C, reuse_a, reuse_b) — f32 flavor: neg_a/neg_b must be 0
    return __builtin_amdgcn_wmma_f32_16x16x4_f32(false, a, false, b, (short)0, c, false, false);
}

constexpr int B_  = 4;
constexpr int S_  = 2048;
constexpr int D_  = 1024;
constexpr int H_  = 16;
constexpr int HD_ = 64;
constexpr float INV_SCALE = 0.125f;       // 1/sqrt(HD)
constexpr int PSTRIDE = 2051;             // LDS score-row stride (odd, conflict-free for both phases)

// ---------------------------------------------------------------------------
// Generic GEMM + bias: Y = X(8192x1024) @ W(1024x1024) + b
// SPLIT_HEADS: write Y as (B,H,S,HD); else plain row-major (8192x1024).
// Block: 128 threads (4 waves), 64x64 output tile, each wave a 32x32 subtile.
// ---------------------------------------------------------------------------
template <bool SPLIT_HEADS>
__global__ __launch_bounds__(128)
void gemm_bias(const float* __restrict__ X, const float* __restrict__ W,
               const float* __restrict__ bias, float* __restrict__ Y)
{
    __shared__ float Xs[64][33];   // pad to 33: conflict-free M-striped A reads
    __shared__ float Ws[32][65];   // pad to 65: de-conflict the two K-halves

    const int lane = threadIdx.x & 31;
    const int wave = threadIdx.x >> 5;
    const int hl   = lane & 15;
    const int koff = (lane < 16) ? 0 : 2;

    const int m0 = blockIdx.y * 64;
    const int n0 = blockIdx.x * 64;
    const int wm = (wave & 1) * 32;
    const int wn = (wave >> 1) * 32;

    v8f c00 = {}, c01 = {}, c10 = {}, c11 = {};

    const int xrow = threadIdx.x >> 3;         // 0..15
    const int xcol = (threadIdx.x & 7) * 4;    // 0..28
    const int wrow = threadIdx.x >> 4;         // 0..7
    const int wcol = (threadIdx.x & 15) * 4;   // 0..60

    for (int k0 = 0; k0 < D_; k0 += 32) {
        __syncthreads();
#pragma unroll
        for (int i = 0; i < 4; ++i) {
            int r = xrow + i * 16;
            float4 x = *(const float4*)(X + (size_t)(m0 + r) * D_ + k0 + xcol);
            Xs[r][xcol + 0] = x.x; Xs[r][xcol + 1] = x.y;
            Xs[r][xcol + 2] = x.z; Xs[r][xcol + 3] = x.w;
        }
#pragma unroll
        for (int i = 0; i < 4; ++i) {
            int r = wrow + i * 8;
            float4 w = *(const float4*)(W + (size_t)(k0 + r) * D_ + n0 + wcol);
            Ws[r][wcol + 0] = w.x; Ws[r][wcol + 1] = w.y;
            Ws[r][wcol + 2] = w.z; Ws[r][wcol + 3] = w.w;
        }
        __syncthreads();

#pragma unroll
        for (int kk = 0; kk < 32; kk += 4) {
            v2f a0 = { Xs[wm + hl][kk + koff],      Xs[wm + hl][kk + koff + 1] };
            v2f a1 = { Xs[wm + 16 + hl][kk + koff], Xs[wm + 16 + hl][kk + koff + 1] };
            v2f b0 = { Ws[kk + koff][wn + hl],      Ws[kk + koff + 1][wn + hl] };
            v2f b1 = { Ws[kk + koff][wn + 16 + hl], Ws[kk + koff + 1][wn + 16 + hl] };
            c00 = wmma_f32(a0, b0, c00);
            c01 = wmma_f32(a0, b1, c01);
            c10 = wmma_f32(a1, b0, c10);
            c11 = wmma_f32(a1, b1, c11);
        }
    }

    const float bias0 = bias[n0 + wn + hl];
    const float bias1 = bias[n0 + wn + 16 + hl];
    const int   mbase = m0 + wm + 8 * (lane >> 4);
    const int   nA    = n0 + wn + hl;
    const int   nB    = nA + 16;

    auto store = [&](int r, int n, float v) {
        if (SPLIT_HEADS) {
            int b = r >> 11, s = r & (S_ - 1);
            int h = n >> 6,  hd = n & (HD_ - 1);
            Y[(((size_t)b * H_ + h) * S_ + s) * HD_ + hd] = v;
        } else {
            Y[(size_t)r * D_ + n] = v;
        }
    };

#pragma unroll
    for (int i = 0; i < 8; ++i) {
        int r0 = mbase + i;
        int r1 = r0 + 16;
        store(r0, nA, c00[i] + bias0);
        store(r0, nB, c01[i] + bias1);
        store(r1, nA, c10[i] + bias0);
        store(r1, nB, c11[i] + bias1);
    }
}

// ---------------------------------------------------------------------------
// Fused attention: per block = (b, h, 16 query rows).
//  1) scores = Q Kᵀ / 8 into LDS stripe [16][PSTRIDE] via WMMA (4 waves split key blocks)
//  2) per-row softmax (wave32 shfl_xor reductions), write normalized attention to d_out
//  3) context = P @ V via WMMA (each wave owns 16 head-dim cols), write (B,S,D) ctx
// Q/K/V are in (B,H,S,HD) layout.  Dynamic LDS: 16*PSTRIDE floats (~131 KB, OK on CDNA5).
// ---------------------------------------------------------------------------
__global__ __launch_bounds__(128)
void attn_fused(const float* __restrict__ Qw, const float* __restrict__ Kw,
                const float* __restrict__ Vw, float* __restrict__ attn,
                float* __restrict__ ctx)
{
    extern __shared__ float P[];   // [16][PSTRIDE]

    const int lane = threadIdx.x & 31;
    const int wave = threadIdx.x >> 5;
    const int hl   = lane & 15;
    const int koff = (lane < 16) ? 0 : 2;

    const int q0 = blockIdx.x * 16;
    const int h  = blockIdx.y;
    const int b  = blockIdx.z;
    const size_t bh = (size_t)b * H_ + h;

    const float* Qb = Qw + (bh * S_ + q0) * HD_;
    const float* Kb = Kw + bh * S_ * HD_;
    const float* Vb = Vw + bh * S_ * HD_;

    // ---- cache Q A-fragments (16 frags cover K=0..63) -----------------------
    v2f aq[16];
#pragma unroll
    for (int j = 0; j < 16; ++j)
        aq[j] = *(const v2f*)(Qb + hl * HD_ + 4 * j + koff);

    // ---- phase 1: scores into LDS ------------------------------------------
    for (int kb = wave; kb < S_ / 16; kb += 4) {
        v8f c = {};
        const float* Kt = Kb + (size_t)(kb * 16 + hl) * HD_;
#pragma unroll
        for (int j = 0; j < 16; ++j) {
            v2f bk = *(const v2f*)(Kt + 4 * j + koff);   // B[k][n] = K[n][k]
            c = wmma_f32(aq[j], bk, c);
        }
        const int mrow = 8 * (lane >> 4);
        const int n    = kb * 16 + hl;
#pragma unroll
        for (int i = 0; i < 8; ++i)
            P[(mrow + i) * PSTRIDE + n] = c[i] * INV_SCALE;
    }
    __syncthreads();

    // ---- phase 2: softmax per row, write attention -------------------------
    for (int r = wave; r < 16; r += 4) {
        float* row = P + r * PSTRIDE;
        float mx = -3.402823466e38f;
        for (int cix = lane; cix < S_; cix += 32) mx = fmaxf(mx, row[cix]);
#pragma unroll
        for (int o = 16; o > 0; o >>= 1) mx = fmaxf(mx, __shfl_xor(mx, o, 32));

        float sum = 0.f;
        for (int cix = lane; cix < S_; cix += 32) {
            float e = __expf(row[cix] - mx);
            row[cix] = e;
            sum += e;
        }
#pragma unroll
        for (int o = 16; o > 0; o >>= 1) sum += __shfl_xor(sum, o, 32);

        const float inv = 1.f / sum;
        float* arow = attn + (bh * S_ + (size_t)(q0 + r)) * S_;
        for (int cix = lane; cix < S_; cix += 32) {
            float p = row[cix] * inv;
            row[cix] = p;        // normalized P stays in LDS for phase 3
            arow[cix] = p;       // attention output (written exactly once)
        }
    }
    __syncthreads();

    // ---- phase 3: context = P @ V (16 x HD), wave owns cols [wave*16, +16) --
    v8f c = {};
    const int ncol = wave * 16 + hl;
#pragma unroll 4
    for (int kk = 0; kk < S_; kk += 4) {
        v2f a  = { P[hl * PSTRIDE + kk + koff], P[hl * PSTRIDE + kk + koff + 1] };
        v2f bv = { Vb[(size_t)(kk + koff) * HD_ + ncol],
                   Vb[(size_t)(kk + koff + 1) * HD_ + ncol] };
        c = wmma_f32(a, bv, c);
    }
    const int mrow = 8 * (lane >> 4);
#pragma unroll
    for (int i = 0; i < 8; ++i) {
        int r   = q0 + mrow + i;
        int col = h * HD_ + wave * 16 + hl;
        ctx[((size_t)b * S_ + r) * D_ + col] = c[i];
    }
}

// ---------------------------------------------------------------------------
// Host-side launcher.
// d_in order: query,key,value,W_q,b_q,W_k,b_k,W_v,b_v,W_o,b_o (all fp32)
// d_out: output (B,S,D) followed by attention (B,H,S,S), fp32
// d_ws:  Q,K,V (B,H,S,HD) + ctx (B,S,D) = 4 * 32 MB = 128 MB
// ---------------------------------------------------------------------------
extern "C" void kernel_launch(void* const* d_in, const int* in_sizes, int n_in,
                              void* d_out, int out_size, void* d_ws, size_t ws_size,
                              hipStream_t stream)
{
    const float* query = (const float*)d_in[0];
    const float* key   = (const float*)d_in[1];
    const float* value = (const float*)d_in[2];
    const float* W_q   = (const float*)d_in[3];
    const float* b_q   = (const float*)d_in[4];
    const float* W_k   = (const float*)d_in[5];
    const float* b_k   = (const float*)d_in[6];
    const float* W_v   = (const float*)d_in[7];
    const float* b_v   = (const float*)d_in[8];
    const float* W_o   = (const float*)d_in[9];
    const float* b_o   = (const float*)d_in[10];

    float* out  = (float*)d_out;                       // (B,S,D)
    float* attn = out + (size_t)B_ * S_ * D_;          // (B,H,S,S)

    const size_t tensor_elems = (size_t)B_ * S_ * D_;  // 8.4M floats = 32 MB
    float* Qws = (float*)d_ws;
    float* Kws = Qws + tensor_elems;
    float* Vws = Kws + tensor_elems;
    float* Ctx = Vws + tensor_elems;

    const dim3 gemm_grid(D_ / 64, (B_ * S_) / 64);     // 16 x 128 blocks
    const dim3 gemm_blk(128);

    // QKV projections -> (B,H,S,HD)
    gemm_bias<true><<<gemm_grid, gemm_blk, 0, stream>>>(query, W_q, b_q, Qws);
    gemm_bias<true><<<gemm_grid, gemm_blk, 0, stream>>>(key,   W_k, b_k, Kws);
    gemm_bias<true><<<gemm_grid, gemm_blk, 0, stream>>>(value, W_v, b_v, Vws);

    // Fused attention (131 KB dynamic LDS per workgroup — within CDNA5's 320 KB)
    const size_t smem = (size_t)16 * PSTRIDE * sizeof(float);
    hipFuncSetAttribute(reinterpret_cast<const void*>(attn_fused),
                        hipFuncAttributeMaxDynamicSharedMemorySize, (int)smem);
    attn_fused<<<dim3(S_ / 16, H_, B_), 128, smem, stream>>>(Qws, Kws, Vws, attn, Ctx);

    // Output projection -> (B,S,D)
    gemm_bias<false><<<gemm_grid, gemm_blk, 0, stream>>>(Ctx, W_o, b_o, out);
}